// GraphEncoderWNNit_40269613367534
// MI455X (gfx1250) — compile-verified
//
#include <hip/hip_runtime.h>
#include <math.h>

typedef float v2f __attribute__((ext_vector_type(2)));
typedef float v8f __attribute__((ext_vector_type(8)));

#define HDIM 64

// ---------------- utility kernels ----------------
__global__ __launch_bounds__(256) void k_fill_zero(float* p, int n) {
    int i = blockIdx.x * 256 + threadIdx.x;
    if (i < n) p[i] = 0.0f;
}

__global__ __launch_bounds__(256) void k_deg_init(float* deg, int n) {
    int i = blockIdx.x * 256 + threadIdx.x;
    if (i < n) deg[i] = 1.0f;   // self-loop
}

__global__ __launch_bounds__(256) void k_deg_accum(const int* __restrict__ dst, float* deg, int e) {
    int i = blockIdx.x * 256 + threadIdx.x;
    if (i < e) atomicAdd(&deg[dst[i]], 1.0f);
}

__global__ __launch_bounds__(256) void k_deg_to_dis(float* deg, int n) {
    int i = blockIdx.x * 256 + threadIdx.x;
    if (i < n) deg[i] = rsqrtf(deg[i]);   // deg >= 1 always (self loops)
}

// async 16B copy global -> LDS (CDNA5, ASYNCcnt-tracked; bypasses VGPRs)
__device__ __forceinline__ void async_copy_b128(const float* gptr, float* lptr) {
    unsigned int       lds_off = (unsigned int)(uintptr_t)lptr;       // low 32 bits = LDS byte addr
    unsigned long long gaddr   = (unsigned long long)(uintptr_t)gptr;
    asm volatile("global_load_async_to_lds_b128 %0, %1, off"
                 :: "v"(lds_off), "v"(gaddr) : "memory");
}
__device__ __forceinline__ void async_wait_all() {
    asm volatile("s_wait_asynccnt 0x0" ::: "memory");
}

// ---------------- WMMA fp32 GEMM: out[N,64] = x[N,K] @ w[K,64] ----------------
// One block = 16 output rows; 4 waves each own a 16-col tile.
// A tile (16x64) and B tile (64x64) double-buffered in LDS via async global->LDS DMA.
__global__ __launch_bounds__(128) void k_gemm_wmma_f32(const float* __restrict__ x,
                                                       const float* __restrict__ w,
                                                       float* __restrict__ out,
                                                       int n, int K) {
    __shared__ float As[2][16 * 64];
    __shared__ float Bs[2][64 * 64];

    const int tid  = threadIdx.x;
    const int wave = tid >> 5;
    const int lane = tid & 31;
    const int half = lane >> 4;   // 0: lanes 0-15, 1: lanes 16-31
    const int l    = lane & 15;
    const int rowBase = blockIdx.x * 16;
    const int nchunks = K >> 6;   // K is a multiple of 64 (512/256/64)

    // issue one K-chunk's async copies into buffer `buf`
    auto issue_chunk = [&](int c, int buf) {
        const int kk0 = c << 6;
        #pragma unroll
        for (int i = 0; i < 2; ++i) {          // A: 16 rows x 64 k = 256 float4
            int f4 = tid + i * 128;
            int r  = f4 >> 4;
            int kl = (f4 & 15) << 2;
            int row = rowBase + r;
            if (row >= n) row = n - 1;          // clamp: those D rows are never stored
            async_copy_b128(x + (size_t)row * K + kk0 + kl, &As[buf][r * 64 + kl]);
        }
        #pragma unroll
        for (int i = 0; i < 8; ++i) {          // B: 64 k-rows x 64 cols = 1024 float4
            int f4 = tid + i * 128;
            int kr = f4 >> 4;
            int kl = (f4 & 15) << 2;
            async_copy_b128(w + (size_t)(kk0 + kr) * HDIM + kl, &Bs[buf][kr * 64 + kl]);
        }
    };

    v8f acc = {0.f, 0.f, 0.f, 0.f, 0.f, 0.f, 0.f, 0.f};

    issue_chunk(0, 0);
    for (int c = 0; c < nchunks; ++c) {
        const int buf = c & 1;
        async_wait_all();        // this wave's outstanding copies (== chunk c) complete
        __syncthreads();         // all waves' copies visible; prior compute on buf done
        if (c + 1 < nchunks) issue_chunk(c + 1, buf ^ 1);   // overlap DMA with WMMAs

        #pragma unroll
        for (int kk = 0; kk < 64; kk += 4) {
            // A 16x4 fragment: lanes 0-15 hold K=kk,kk+1; lanes 16-31 hold K=kk+2,kk+3
            v2f a, b;
            int kA = kk + half * 2;
            a.x = As[buf][l * 64 + kA];
            a.y = As[buf][l * 64 + kA + 1];
            // B 4x16 fragment: VGPR0 = K rows {kk, kk+2} by lane half, VGPR1 = {kk+1, kk+3}
            int col = wave * 16 + l;
            b.x = Bs[buf][kA * 64 + col];
            b.y = Bs[buf][(kA + 1) * 64 + col];
            acc = __builtin_amdgcn_wmma_f32_16x16x4_f32(false, a, false, b,
                                                        (short)0, acc, false, false);
        }
        __syncthreads();         // done reading buf before chunk c+2 overwrites it
    }

    // C/D layout: VGPR v -> (M=v, lanes 0-15) and (M=v+8, lanes 16-31)
    int col = wave * 16 + l;
    #pragma unroll
    for (int v = 0; v < 8; ++v) {
        int row = rowBase + v + half * 8;
        if (row < n) out[(size_t)row * HDIM + col] = acc[v];
    }
}

// ---------------- edge scatter: agg[dst] += t[src] * dis[src]*dis[dst] ----------------
__global__ __launch_bounds__(256) void k_scatter_edges(const float* __restrict__ t,
                                                       const int* __restrict__ src,
                                                       const int* __restrict__ dst,
                                                       const float* __restrict__ dis,
                                                       float* agg, int e) {
    size_t idx = (size_t)blockIdx.x * 256 + threadIdx.x;
    int ei = (int)(idx >> 6);
    int j  = (int)(idx & 63);
    if (ei < e) {
        int s = src[ei];
        int d = dst[ei];
        float nrm = dis[s] * dis[d];
        atomicAdd(&agg[(size_t)d * HDIM + j], t[(size_t)s * HDIM + j] * nrm);
    }
}

// ---------------- epilogue: self-loop + bias + PReLU ----------------
__global__ __launch_bounds__(256) void k_post_prelu(const float* agg, const float* t,
                                                    const float* dis, const float* b,
                                                    const float* a, float* out, int n) {
    size_t idx = (size_t)blockIdx.x * 256 + threadIdx.x;
    if (idx < (size_t)n * HDIM) {
        int i = (int)(idx >> 6);
        int j = (int)(idx & 63);
        float d = dis[i];
        float v = agg[idx] + t[idx] * d * d + b[j];
        out[idx] = (v >= 0.f) ? v : v * a[j];
    }
}

// ---------------- epilogue with L2 row-norm: one wave32 per row ----------------
__global__ __launch_bounds__(256) void k_post_prelu_l2(const float* agg, const float* t,
                                                       const float* dis, const float* b,
                                                       const float* a, float* out, int n) {
    int row  = blockIdx.x * 8 + (threadIdx.x >> 5);
    int lane = threadIdx.x & 31;
    if (row >= n) return;
    float d2 = dis[row] * dis[row];
    size_t base = (size_t)row * HDIM;
    int j0 = lane, j1 = lane + 32;

    float v0 = agg[base + j0] + t[base + j0] * d2 + b[j0];
    v0 = (v0 >= 0.f) ? v0 : v0 * a[j0];
    float v1 = agg[base + j1] + t[base + j1] * d2 + b[j1];
    v1 = (v1 >= 0.f) ? v1 : v1 * a[j1];

    float ss = v0 * v0 + v1 * v1;
    #pragma unroll
    for (int off = 16; off > 0; off >>= 1)
        ss += __shfl_xor(ss, off, 32);
    float scale = 1.0f / fmaxf(sqrtf(ss), 1e-12f);
    out[base + j0] = v0 * scale;
    out[base + j1] = v1 * scale;
}

// ---------------- final combine & tail copies ----------------
__global__ __launch_bounds__(256) void k_combine(const float* h1, const float* h2,
                                                 const float* __restrict__ w1,
                                                 const float* __restrict__ w2,
                                                 float* out, int n) {
    size_t idx = (size_t)blockIdx.x * 256 + threadIdx.x;
    if (idx < (size_t)n * HDIM) {
        int i = (int)(idx >> 6);
        out[idx] = h1[idx] * w1[i] + h2[idx] * w2[i];
    }
}

__global__ __launch_bounds__(256) void k_copy_w(const float* __restrict__ w1,
                                                const float* __restrict__ w2,
                                                float* o1, float* o2, int n) {
    int i = blockIdx.x * 256 + threadIdx.x;
    if (i < n) { o1[i] = w1[i]; o2[i] = w2[i]; }
}

static inline int cdiv_ll(long long a, long long b) { return (int)((a + b - 1) / b); }

extern "C" void kernel_launch(void* const* d_in, const int* in_sizes, int n_in,
                              void* d_out, int out_size, void* d_ws, size_t ws_size,
                              hipStream_t stream) {
    const float* x1 = (const float*)d_in[0];
    const float* x2 = (const float*)d_in[1];
    const int*   ei = (const int*)d_in[2];
    const float* w1 = (const float*)d_in[4];
    const float* w2 = (const float*)d_in[5];
    const float* W1 = (const float*)d_in[6];
    const float* b1 = (const float*)d_in[7];
    const float* W2 = (const float*)d_in[8];
    const float* b2 = (const float*)d_in[9];
    const float* W3 = (const float*)d_in[10];
    const float* b3 = (const float*)d_in[11];
    const float* W4 = (const float*)d_in[12];
    const float* b4 = (const float*)d_in[13];
    const float* a1 = (const float*)d_in[14];
    const float* a2 = (const float*)d_in[15];
    const float* a3 = (const float*)d_in[16];
    const float* a4 = (const float*)d_in[17];

    const int N   = in_sizes[4];          // w1 has N elements
    const int E   = in_sizes[2] / 2;
    const int IN1 = in_sizes[0] / N;      // 512
    const int IN2 = in_sizes[1] / N;      // 256
    const int* src = ei;
    const int* dst = ei + E;

    // workspace: dis[N], B0[N*64], B1[N*64]; d_out's first N*64 doubles as h1 buffer
    float* dis = (float*)d_ws;
    float* B0  = dis + N;
    float* B1  = B0 + (size_t)N * HDIM;
    float* out   = (float*)d_out;
    float* outw1 = out + (size_t)N * HDIM;
    float* outw2 = outw1 + N;

    const int NH  = N * HDIM;
    const int gN  = cdiv_ll(N, 256);
    const int gE  = cdiv_ll(E, 256);
    const int gNH = cdiv_ll((long long)N * HDIM, 256);
    const int gEH = cdiv_ll((long long)E * HDIM, 256);
    const int gGemm = cdiv_ll(N, 16);
    const int gRow  = cdiv_ll(N, 8);

    // symmetric GCN normalization (with self-loops)
    k_deg_init<<<gN, 256, 0, stream>>>(dis, N);
    k_deg_accum<<<gE, 256, 0, stream>>>(dst, dis, E);
    k_deg_to_dis<<<gN, 256, 0, stream>>>(dis, N);

    // ---- branch 1, layer 1: prelu(gcn(x1, W1, b1), a1) -> B0
    k_gemm_wmma_f32<<<gGemm, 128, 0, stream>>>(x1, W1, B0, N, IN1);
    k_fill_zero<<<gNH, 256, 0, stream>>>(B1, NH);
    k_scatter_edges<<<gEH, 256, 0, stream>>>(B0, src, dst, dis, B1, E);
    k_post_prelu<<<gNH, 256, 0, stream>>>(B1, B0, dis, b1, a1, B0, N);
    // ---- branch 1, layer 2: l2norm(prelu(gcn(B0, W3, b3), a2)) -> out (h1)
    k_gemm_wmma_f32<<<gGemm, 128, 0, stream>>>(B0, W3, B1, N, HDIM);
    k_fill_zero<<<gNH, 256, 0, stream>>>(out, NH);
    k_scatter_edges<<<gEH, 256, 0, stream>>>(B1, src, dst, dis, out, E);
    k_post_prelu_l2<<<gRow, 256, 0, stream>>>(out, B1, dis, b3, a2, out, N);

    // ---- branch 2, layer 1: prelu(gcn(x2, W2, b2), a3) -> B0
    k_gemm_wmma_f32<<<gGemm, 128, 0, stream>>>(x2, W2, B0, N, IN2);
    k_fill_zero<<<gNH, 256, 0, stream>>>(B1, NH);
    k_scatter_edges<<<gEH, 256, 0, stream>>>(B0, src, dst, dis, B1, E);
    k_post_prelu<<<gNH, 256, 0, stream>>>(B1, B0, dis, b2, a3, B0, N);
    // ---- branch 2, layer 2: l2norm(prelu(gcn(B0, W4, b4), a4)) -> B0 (h2)
    k_gemm_wmma_f32<<<gGemm, 128, 0, stream>>>(B0, W4, B1, N, HDIM);
    k_fill_zero<<<gNH, 256, 0, stream>>>(B0, NH);
    k_scatter_edges<<<gEH, 256, 0, stream>>>(B1, src, dst, dis, B0, E);
    k_post_prelu_l2<<<gRow, 256, 0, stream>>>(B0, B1, dis, b4, a4, B0, N);

    // ---- combine: out = h1*w1 + h2*w2 ; append w1, w2
    k_combine<<<gNH, 256, 0, stream>>>(out, B0, w1, w2, out, N);
    k_copy_w<<<gN, 256, 0, stream>>>(w1, w2, outw1, outw2, N);
}